// FashionNetLoss_26783416058097
// MI455X (gfx1250) — compile-verified
//
#include <hip/hip_runtime.h>
#include <math.h>

// ---------------------------------------------------------------------------
// FashionNet YOLO-style loss for MI455X (gfx1250, wave32).
// Decomposition: loss_obj = sum focal(x,0) over obj planes  (+ per-positive-cell
// correction), loss_box/loss_cls only touch the 512 positive cells per scale.
// Reductions use V_WMMA_F32_16X16X4_F32 with B == ones as an exact f32
// wave-wide summation unit.
// ---------------------------------------------------------------------------

typedef float v2f __attribute__((ext_vector_type(2)));
typedef float v8f __attribute__((ext_vector_type(8)));

#define EPSF 1e-7f
#define FOUR_OVER_PI2 0.40528473456935109f

// ---- workspace layout (floats) --------------------------------------------
// [0,512)       : scale0 obj-plane block partials (512 blocks)
// [512,640)     : scale1 obj-plane block partials (128 blocks)
// [640,672)     : scale2 obj-plane block partials (32 blocks)
// [672,672+4608): per (scale,target) triples {box, obj_corr, cls_sum}
#define NEG_OFF0 0
#define NEG_OFF1 512
#define NEG_OFF2 640
#define TGT_OFF  672
#define NB0 512
#define NB1 128
#define NB2 32

// ---- exact f32 wave-wide sum via WMMA -------------------------------------
// A (16x4 f32) holds 64 values: lane l contributes {a0, a1}. B = ones(4x16).
// D[m][n] = rowsum(m); summing one lane's 8 D regs + xor-16 partner gives the
// grand total (sum of all 64 A elements) in every lane, in full f32 precision.
__device__ __forceinline__ float wave_sum(float a0, float a1) {
  v2f A; A[0] = a0; A[1] = a1;
  v2f Bm; Bm[0] = 1.0f; Bm[1] = 1.0f;
  v8f C = {};
  C = __builtin_amdgcn_wmma_f32_16x16x4_f32(
      /*neg_a=*/false, A, /*neg_b=*/false, Bm,
      /*c_mod=*/(short)0, C, /*reuse_a=*/false, /*reuse_b=*/false);
  float r = ((C[0] + C[1]) + (C[2] + C[3])) + ((C[4] + C[5]) + (C[6] + C[7]));
  r += __shfl_xor(r, 16, 32);
  return r;  // full wave sum in every lane
}

// Deterministic 256-thread block sum (valid in thread 0). Must be called by
// all 256 threads (WMMA needs EXEC all-ones).
__device__ __forceinline__ float block_sum_256(float v, float* lds) {
  float w = wave_sum(v, 0.0f);
  int lane = threadIdx.x & 31, wid = threadIdx.x >> 5;
  if (lane == 0) lds[wid] = w;
  __syncthreads();
  float r = (threadIdx.x < 8) ? lds[threadIdx.x] : 0.0f;
  r += __shfl_xor(r, 4, 32);
  r += __shfl_xor(r, 2, 32);
  r += __shfl_xor(r, 1, 32);
  __syncthreads();
  return r;
}

// ---- loss math (mirrors reference numerics) -------------------------------
__device__ __forceinline__ float bce_logits(float x, float t) {
  return fmaxf(x, 0.0f) - x * t + log1pf(expf(-fabsf(x)));
}
__device__ __forceinline__ float focal_elem(float x, float t) {
  float b  = bce_logits(x, t);
  float pt = expf(-b);
  float om = 1.0f - pt;                 // >= 0 since b >= 0
  return 0.25f * (om * sqrtf(om)) * b;  // alpha * (1-pt)^1.5 * bce
}

// ---- kernel 1: sum focal(x,0) over the channel-4 (objectness) plane -------
__global__ __launch_bounds__(256) void obj_neg_kernel(
    const float* __restrict__ p, int gs2_4, int gs2, int nf4,
    float* __restrict__ partial) {
  __shared__ float lds[8];
  float a0 = 0.0f, a1 = 0.0f;
  int stride = gridDim.x * blockDim.x;
  for (int i = blockIdx.x * blockDim.x + threadIdx.x; i < nf4; i += stride) {
    int b = i / gs2_4;
    int r = i - b * gs2_4;
    const float4* q =
        reinterpret_cast<const float4*>(p + (size_t)b * 18 * gs2 + (size_t)4 * gs2) + r;
    __builtin_prefetch(reinterpret_cast<const char*>(q) + 4096, 0, 0);
    float4 v = *q;
    a0 += focal_elem(v.x, 0.0f) + focal_elem(v.z, 0.0f);
    a1 += focal_elem(v.y, 0.0f) + focal_elem(v.w, 0.0f);
  }
  float w = wave_sum(a0, a1);
  int lane = threadIdx.x & 31, wid = threadIdx.x >> 5;
  if (lane == 0) lds[wid] = w;
  __syncthreads();
  if (threadIdx.x < 32) {
    float r = (threadIdx.x < 8) ? lds[threadIdx.x] : 0.0f;
    r += __shfl_xor(r, 4, 32);
    r += __shfl_xor(r, 2, 32);
    r += __shfl_xor(r, 1, 32);
    if (threadIdx.x == 0) partial[blockIdx.x] = r;
  }
}

// ---- kernel 2: per-(scale,target) contributions ---------------------------
__global__ __launch_bounds__(256) void target_kernel(
    const float* __restrict__ p0, const float* __restrict__ p1,
    const float* __restrict__ p2, const float* __restrict__ tgt,
    float* __restrict__ out) {
  int idx = blockIdx.x * blockDim.x + threadIdx.x;
  if (idx >= 3 * 512) return;
  int s = idx / 512, k = idx - s * 512;
  int gs = (s == 0) ? 160 : ((s == 1) ? 80 : 40);
  const float* p = (s == 0) ? p0 : ((s == 1) ? p1 : p2);
  const float* T = tgt + (size_t)k * 6;

  int   bi  = (int)T[0];
  int   cls = (int)T[1];
  float fgs = (float)gs;
  float cx = T[2] * fgs, cy = T[3] * fgs, w = T[4] * fgs, h = T[5] * fgs;
  int gi = (int)floorf(cx); gi = max(0, min(gi, gs - 1));
  int gj = (int)floorf(cy); gj = max(0, min(gj, gs - 1));
  int gs2 = gs * gs;
  const float* base = p + (size_t)bi * 18 * gs2 + (size_t)gj * gs + gi;

  float lx  = base[0];
  float ly  = base[(size_t)gs2];
  float pw  = base[(size_t)2 * gs2];
  float ph  = base[(size_t)3 * gs2];
  float obj = base[(size_t)4 * gs2];

  float px = 1.0f / (1.0f + expf(-lx));
  float py = 1.0f / (1.0f + expf(-ly));
  float tx = cx - (float)gi, ty = cy - (float)gj;

  // CIoU (identical formula to reference; pw/ph are raw logits)
  float px1 = px - pw * 0.5f, px2 = px + pw * 0.5f;
  float py1 = py - ph * 0.5f, py2 = py + ph * 0.5f;
  float tx1 = tx - w * 0.5f,  tx2 = tx + w * 0.5f;
  float ty1 = ty - h * 0.5f,  ty2 = ty + h * 0.5f;
  float iw = fmaxf(fminf(px2, tx2) - fmaxf(px1, tx1), 0.0f);
  float ih = fmaxf(fminf(py2, ty2) - fmaxf(py1, ty1), 0.0f);
  float inter = iw * ih;
  float uni   = pw * ph + w * h - inter + EPSF;
  float iou   = inter / uni;
  float ew = fmaxf(fmaxf(px2, tx2) - fminf(px1, tx1), 0.0f);
  float eh = fmaxf(fmaxf(py2, ty2) - fminf(py1, ty1), 0.0f);
  float c2   = ew * ew + eh * eh + EPSF;
  float rho2 = (px - tx) * (px - tx) + (py - ty) * (py - ty);
  float da = atanf(w / (h + EPSF)) - atanf(pw / (ph + EPSF));
  float v  = FOUR_OVER_PI2 * da * da;
  float alpha = v / (1.0f - iou + v + EPSF);
  float ciou  = iou - (rho2 / c2 + v * alpha);

  float boxl  = 1.0f - ciou;
  float t_obj = fmaxf(ciou, 0.0f);
  float objc  = focal_elem(obj, t_obj) - focal_elem(obj, 0.0f);

  float clssum = 0.0f;
#pragma unroll
  for (int c = 0; c < 13; ++c) {
    float logit = base[(size_t)(5 + c) * gs2];
    float tc = (c == cls) ? 0.95f : 0.0f;
    clssum += bce_logits(logit, tc);
  }

  float* o = out + (size_t)idx * 3;
  o[0] = boxl; o[1] = objc; o[2] = clssum;
}

// ---- kernel 3: deterministic finalize -------------------------------------
__global__ __launch_bounds__(256) void finalize_kernel(
    const float* __restrict__ ws, float* __restrict__ out) {
  __shared__ float lds[8];
  const int nb[3]  = {NB0, NB1, NB2};
  const int off[3] = {NEG_OFF0, NEG_OFF1, NEG_OFF2};
  float res[12];

  for (int s = 0; s < 3; ++s) {
    float a = 0.0f;
    for (int i = threadIdx.x; i < nb[s]; i += 256) a += ws[off[s] + i];
    res[s] = block_sum_256(a, lds);
  }
  const float* tg = ws + TGT_OFF;
  for (int s = 0; s < 3; ++s) {
    float ab = 0.0f, ao = 0.0f, ac = 0.0f;
    for (int k = threadIdx.x; k < 512; k += 256) {
      const float* o = tg + (size_t)(s * 512 + k) * 3;
      ab += o[0]; ao += o[1]; ac += o[2];
    }
    res[3 + s * 3 + 0] = block_sum_256(ab, lds);
    res[3 + s * 3 + 1] = block_sum_256(ao, lds);
    res[3 + s * 3 + 2] = block_sum_256(ac, lds);
  }
  if (threadIdx.x == 0) {
    const float ncell[3] = {32.0f * 160 * 160, 32.0f * 80 * 80, 32.0f * 40 * 40};
    float loss_box = 0.0f, loss_obj = 0.0f, loss_cls = 0.0f;
    for (int s = 0; s < 3; ++s) {
      loss_obj += (res[s] + res[3 + s * 3 + 1]) / ncell[s];
      loss_box += res[3 + s * 3 + 0] / 512.0f;
      loss_cls += res[3 + s * 3 + 2] / (512.0f * 13.0f);
    }
    float total = 5.0f * loss_box + 1.0f * loss_obj + 0.5f * loss_cls;
    out[0] = total; out[1] = loss_box; out[2] = loss_obj; out[3] = loss_cls;
  }
}

// ---------------------------------------------------------------------------
extern "C" void kernel_launch(void* const* d_in, const int* in_sizes, int n_in,
                              void* d_out, int out_size, void* d_ws, size_t ws_size,
                              hipStream_t stream) {
  (void)in_sizes; (void)n_in; (void)out_size; (void)ws_size;
  const float* p0  = (const float*)d_in[0];
  const float* p1  = (const float*)d_in[1];
  const float* p2  = (const float*)d_in[2];
  const float* tgt = (const float*)d_in[3];
  float* ws  = (float*)d_ws;
  float* out = (float*)d_out;

  // Objectness-plane streaming passes (channel 4 only: ~4.3 MB total traffic).
  obj_neg_kernel<<<NB0, 256, 0, stream>>>(p0, 160 * 160 / 4, 160 * 160,
                                          32 * 160 * 160 / 4, ws + NEG_OFF0);
  obj_neg_kernel<<<NB1, 256, 0, stream>>>(p1, 80 * 80 / 4, 80 * 80,
                                          32 * 80 * 80 / 4, ws + NEG_OFF1);
  obj_neg_kernel<<<NB2, 256, 0, stream>>>(p2, 40 * 40 / 4, 40 * 40,
                                          32 * 40 * 40 / 4, ws + NEG_OFF2);
  // 1536 positive-cell evaluations (box CIoU, obj correction, class BCE).
  target_kernel<<<6, 256, 0, stream>>>(p0, p1, p2, tgt, ws + TGT_OFF);
  // Deterministic reduction + weighted combine.
  finalize_kernel<<<1, 256, 0, stream>>>(ws, out);
}